// GNN_69441031242453
// MI455X (gfx1250) — compile-verified
//
#include <hip/hip_runtime.h>
#include <hip/hip_bf16.h>

#define N_NODES   50000
#define HC        128          // H*C for both layers
#define SLOPE_ATT 0.2f
#define SLOPE_ACT 0.01f

typedef __attribute__((ext_vector_type(2))) float v2f;
typedef __attribute__((ext_vector_type(8))) float v8f;

__device__ __forceinline__ float leaky(float x, float s) { return x >= 0.f ? x : s * x; }

// Order-preserving float -> uint key (for atomicMax-based segment max).
// key(0u) acts as "below -inf"; every node has a self-loop so m is always set.
__device__ __forceinline__ unsigned fkey(float v) {
    unsigned b = __float_as_uint(v);
    return (b & 0x80000000u) ? ~b : (b | 0x80000000u);
}
__device__ __forceinline__ float funkey(unsigned k) {
    return __uint_as_float((k & 0x80000000u) ? (k & 0x7FFFFFFFu) : ~k);
}

// ---------------------------------------------------------------------------
// GEMM: C[M x 128] = A[M x K] * B[K x 128], fp32, via V_WMMA_F32_16X16X4_F32.
// One block = 256 threads = 8 waves; block covers 16 rows, wave w covers cols
// [16w, 16w+16). A tile staged in LDS with padded stride to avoid conflicts.
// ---------------------------------------------------------------------------
template <int K>
__global__ __launch_bounds__(256) void gemm_wmma_f32(const float* __restrict__ A,
                                                     const float* __restrict__ B,
                                                     float* __restrict__ C) {
    constexpr int LDA = K + 1;
    __shared__ float As[16 * LDA];

    const int row_base = blockIdx.x * 16;

    // Cooperative load of the 16 x K A-tile into LDS.
    for (int i = threadIdx.x; i < 16 * K; i += 256) {
        const int r = i / K;
        const int c = i - r * K;
        As[r * LDA + c] = A[(row_base + r) * K + c];
    }
    __syncthreads();

    const int lane = threadIdx.x & 31;
    const int wave = threadIdx.x >> 5;
    const int m    = lane & 15;   // A row / B col within tile
    const int g    = lane >> 4;   // half-wave group
    const int col  = wave * 16 + m;

    v8f acc = {};
    for (int k = 0; k < K; k += 4) {
        v2f a, b;
        // 32-bit A 16x4 layout: lanes 0-15 K={0,1}, lanes 16-31 K={2,3}
        a.x = As[m * LDA + k + g * 2 + 0];
        a.y = As[m * LDA + k + g * 2 + 1];
        // 32-bit B 4x16 layout (symmetric): VGPR v, lane-group g -> K = 2g+v
        b.x = B[(k + g * 2 + 0) * HC + col];
        b.y = B[(k + g * 2 + 1) * HC + col];
        acc = __builtin_amdgcn_wmma_f32_16x16x4_f32(false, a, false, b,
                                                    (short)0, acc, false, false);
    }

    // C/D layout: VGPR v, lanes 0-15 -> M=v, lanes 16-31 -> M=v+8, N=lane&15.
#pragma unroll
    for (int v = 0; v < 8; ++v)
        C[(row_base + g * 8 + v) * HC + col] = acc[v];
}

// ---------------------------------------------------------------------------
// Per-node attention scores: a_src[n,h] = sum_c h[n,h,c]*att_src[h,c], same dst
// ---------------------------------------------------------------------------
template <int H>
__global__ void att_scores(const float* __restrict__ h,
                           const float* __restrict__ att_src,
                           const float* __restrict__ att_dst,
                           float* __restrict__ a_src, float* __restrict__ a_dst) {
    constexpr int C = HC / H;
    const int idx = blockIdx.x * blockDim.x + threadIdx.x;
    if (idx >= N_NODES * H) return;
    const int n  = idx / H;
    const int hh = idx - n * H;
    const float* hp = h + n * HC + hh * C;
    float ss = 0.f, ds = 0.f;
#pragma unroll 4
    for (int c = 0; c < C; ++c) {
        const float v = hp[c];
        ss += v * att_src[hh * C + c];
        ds += v * att_dst[hh * C + c];
    }
    a_src[idx] = ss;
    a_dst[idx] = ds;
}

// ---------------------------------------------------------------------------
// Edge pass 1: e = leaky(a_src[src]+a_dst[dst]); segment max via uint keys.
// ---------------------------------------------------------------------------
template <int H>
__global__ void edge_max(const long long* __restrict__ es, const long long* __restrict__ ed,
                         int E, int ET,
                         const float* __restrict__ a_src, const float* __restrict__ a_dst,
                         float* __restrict__ ebuf, unsigned* __restrict__ mkey) {
    const int tid = blockIdx.x * blockDim.x + threadIdx.x;
    if (tid >= ET * H) return;
    const int e  = tid / H;
    const int hh = tid - e * H;
    const int s  = (e < E) ? (int)es[e] : (e - E);
    const int d  = (e < E) ? (int)ed[e] : (e - E);
    const float v = leaky(a_src[s * H + hh] + a_dst[d * H + hh], SLOPE_ATT);
    ebuf[tid] = v;
    atomicMax(&mkey[d * H + hh], fkey(v));
}

// ---------------------------------------------------------------------------
// Edge pass 2: ex = exp(e - m[dst]); Z[dst] += ex (overwrite ebuf with ex).
// ---------------------------------------------------------------------------
template <int H>
__global__ void edge_exp(const long long* __restrict__ es, const long long* __restrict__ ed,
                         int E, int ET,
                         float* __restrict__ ebuf, const unsigned* __restrict__ mkey,
                         float* __restrict__ Z) {
    const int tid = blockIdx.x * blockDim.x + threadIdx.x;
    if (tid >= ET * H) return;
    const int e  = tid / H;
    const int hh = tid - e * H;
    const int d  = (e < E) ? (int)ed[e] : (e - E);
    const float mm = funkey(mkey[d * H + hh]);
    const float ex = __expf(ebuf[tid] - mm);
    ebuf[tid] = ex;
    atomicAdd(&Z[d * H + hh], ex);
}

// ---------------------------------------------------------------------------
// Edge pass 3: out[dst] += h[src] * alpha. 32 threads/edge, float4 per thread.
// ---------------------------------------------------------------------------
template <int H>
__global__ void edge_agg(const long long* __restrict__ es, const long long* __restrict__ ed,
                         int E, int ET,
                         const float* __restrict__ ebuf, const float* __restrict__ Z,
                         const float* __restrict__ h, float* __restrict__ out) {
    constexpr int C = HC / H;
    const int tid = blockIdx.x * blockDim.x + threadIdx.x;
    if (tid >= ET * 32) return;
    const int e  = tid >> 5;
    const int f  = (tid & 31) * 4;     // feature base within [0,128)
    const int hh = f / C;
    const int s  = (e < E) ? (int)es[e] : (e - E);
    const int d  = (e < E) ? (int)ed[e] : (e - E);
    const float alpha = ebuf[e * H + hh] / (Z[d * H + hh] + 1e-16f);
    const float4 hv = *(const float4*)(h + s * HC + f);
    float* op = out + d * HC + f;
    atomicAdd(op + 0, hv.x * alpha);
    atomicAdd(op + 1, hv.y * alpha);
    atomicAdd(op + 2, hv.z * alpha);
    atomicAdd(op + 3, hv.w * alpha);
}

// ---------------------------------------------------------------------------
// Bias + leaky activation, in place.
// ---------------------------------------------------------------------------
__global__ void bias_act(float* __restrict__ x, const float* __restrict__ b, int total) {
    const int tid = blockIdx.x * blockDim.x + threadIdx.x;
    if (tid >= total) return;
    x[tid] = leaky(x[tid] + b[tid & (HC - 1)], SLOPE_ACT);
}

// ---------------------------------------------------------------------------
extern "C" void kernel_launch(void* const* d_in, const int* in_sizes, int n_in,
                              void* d_out, int out_size, void* d_ws, size_t ws_size,
                              hipStream_t stream) {
    const float*     x    = (const float*)d_in[0];
    const long long* eidx = (const long long*)d_in[1];   // int64 [2, E]
    const float*     W1   = (const float*)d_in[2];
    const float*     as1  = (const float*)d_in[3];
    const float*     ad1  = (const float*)d_in[4];
    const float*     b1   = (const float*)d_in[5];
    const float*     W2   = (const float*)d_in[6];
    const float*     as2  = (const float*)d_in[7];
    const float*     ad2  = (const float*)d_in[8];
    const float*     b2   = (const float*)d_in[9];
    float*           out  = (float*)d_out;

    const int E  = in_sizes[1] / 2;      // 800000
    const int ET = E + N_NODES;          // + self loops
    const long long* es = eidx;
    const long long* ed = eidx + E;

    // Workspace layout (floats)
    float*    hbuf = (float*)d_ws;                  // [N*128] GEMM out (both layers)
    float*    out1 = hbuf + (size_t)N_NODES * HC;   // [N*128] layer-1 aggregate
    float*    a_s  = out1 + (size_t)N_NODES * HC;   // [N*4]
    float*    a_d  = a_s + N_NODES * 4;             // [N*4]
    unsigned* mkey = (unsigned*)(a_d + N_NODES * 4);// [N*4]
    float*    Zb   = (float*)(mkey + N_NODES * 4);  // [N*4]
    float*    ebuf = Zb + N_NODES * 4;              // [ET*4]

    const int TB = 256;
    const int gemm_blocks = N_NODES / 16;           // 50000/16 = 3125 exactly

    // ================= Layer 1 (H=4, C=32, K=256) =================
    gemm_wmma_f32<256><<<gemm_blocks, TB, 0, stream>>>(x, W1, hbuf);
    att_scores<4><<<(N_NODES * 4 + TB - 1) / TB, TB, 0, stream>>>(hbuf, as1, ad1, a_s, a_d);

    hipMemsetAsync(mkey, 0, (size_t)N_NODES * 4 * sizeof(unsigned), stream);
    hipMemsetAsync(Zb,   0, (size_t)N_NODES * 4 * sizeof(float), stream);
    hipMemsetAsync(out1, 0, (size_t)N_NODES * HC * sizeof(float), stream);

    edge_max<4><<<((size_t)ET * 4 + TB - 1) / TB, TB, 0, stream>>>(es, ed, E, ET, a_s, a_d, ebuf, mkey);
    edge_exp<4><<<((size_t)ET * 4 + TB - 1) / TB, TB, 0, stream>>>(es, ed, E, ET, ebuf, mkey, Zb);
    edge_agg<4><<<((size_t)ET * 32 + TB - 1) / TB, TB, 0, stream>>>(es, ed, E, ET, ebuf, Zb, hbuf, out1);
    bias_act<<<(N_NODES * HC + TB - 1) / TB, TB, 0, stream>>>(out1, b1, N_NODES * HC);

    // ================= Layer 2 (H=1, C=128, K=128) =================
    gemm_wmma_f32<128><<<gemm_blocks, TB, 0, stream>>>(out1, W2, hbuf);
    att_scores<1><<<(N_NODES + TB - 1) / TB, TB, 0, stream>>>(hbuf, as2, ad2, a_s, a_d);

    hipMemsetAsync(mkey, 0, (size_t)N_NODES * sizeof(unsigned), stream);
    hipMemsetAsync(Zb,   0, (size_t)N_NODES * sizeof(float), stream);
    hipMemsetAsync(out,  0, (size_t)N_NODES * HC * sizeof(float), stream);

    edge_max<1><<<((size_t)ET + TB - 1) / TB, TB, 0, stream>>>(es, ed, E, ET, a_s, a_d, ebuf, mkey);
    edge_exp<1><<<((size_t)ET + TB - 1) / TB, TB, 0, stream>>>(es, ed, E, ET, ebuf, mkey, Zb);
    edge_agg<1><<<((size_t)ET * 32 + TB - 1) / TB, TB, 0, stream>>>(es, ed, E, ET, ebuf, Zb, hbuf, out);
    bias_act<<<(N_NODES * HC + TB - 1) / TB, TB, 0, stream>>>(out, b2, N_NODES * HC);
}